// IrrepsConvolutionBlock_64742337020473
// MI455X (gfx1250) — compile-verified
//
#include <hip/hip_runtime.h>
#include <math.h>

#define MULCH 32
#define NNODES 32768
#define NEDGES 524288
#define RBF_DIM 8
#define MID_DIM 352
#define W_NUMEL 160
#define NODE_DIM 128

#define WAVES_PER_BLOCK 4
#define EDGES_PER_WAVE 16
#define HPAD 68    /* 64 + 4 pad: kills LDS bank conflicts on column reads   */
#define WPAD 164   /* 160 + 4 pad: kills conflicts on the D-layout stores    */

typedef __attribute__((ext_vector_type(2))) float v2f;
typedef __attribute__((ext_vector_type(8))) float v8f;

#define INV_SQRT3F 0.57735026918962576451f
#define INV_SQRT2F 0.70710678118654752440f

// CDNA5 has a hardware V_TANH_F32 (TRANS32 pipe). Prefer it; otherwise use a
// branch-free exp2-based tanh.
__device__ __forceinline__ float fast_tanh(float x) {
#if __has_builtin(__builtin_amdgcn_tanhf)
    return __builtin_amdgcn_tanhf(x);
#elif __has_builtin(__builtin_amdgcn_tanh_f32)
    return __builtin_amdgcn_tanh_f32(x);
#else
    // tanh(x) = sign(x) * (1 - 2/(exp2(2*|x|*log2(e)) + 1))
    float ax = __builtin_fabsf(x);
    float t  = __builtin_amdgcn_exp2f(ax * 2.885390081777926774f); // 2*log2(e)
    float r  = 1.0f - 2.0f / (t + 1.0f);
    return __builtin_copysignf(r, x);
#endif
}

__global__ void zero_out_region(float4* __restrict__ p, int n4) {
    int i = blockIdx.x * blockDim.x + threadIdx.x;
    if (i < n4) p[i] = make_float4(0.f, 0.f, 0.f, 0.f);
}

__global__ __launch_bounds__(128)
void irreps_conv_fused(const float* __restrict__ node,   // (NNODES,128)
                       const float* __restrict__ esph,   // (E,4)
                       const float* __restrict__ erbf,   // (E,8)
                       const int*   __restrict__ eidx,   // (E,2)
                       const float* __restrict__ W1,     // (8,64)
                       const float* __restrict__ W2,     // (64,64)
                       const float* __restrict__ W3,     // (64,160)
                       const float* __restrict__ denom,  // (1,)
                       float* __restrict__ out,          // (NNODES,352)
                       float* __restrict__ msg)          // (E,352)
{
    __shared__ float h_lds[WAVES_PER_BLOCK][16][HPAD];
    __shared__ float w_lds[WAVES_PER_BLOCK][16][WPAD];

    const int tid  = threadIdx.x;
    const int wave = tid >> 5;
    const int lane = tid & 31;
    const int half = lane >> 4;    // 0: K=0,1 side ; 1: K=2,3 side
    const int mrow = lane & 15;    // M row (A) / N col (B,C,D)
    const int kk   = half * 2;

    const int e0 = blockIdx.x * (WAVES_PER_BLOCK * EDGES_PER_WAVE)
                 + wave * EDGES_PER_WAVE;

    const float s1  = 0.35355339059327373f;  // 1/sqrt(8)
    const float s23 = 0.125f;                // 1/sqrt(64)

    // ---------------- Layer 1: (16x8) @ (8x64), tanh -> h_lds -------------
    {
        const float* arow = erbf + (size_t)(e0 + mrow) * RBF_DIM;
#pragma unroll
        for (int t = 0; t < 4; ++t) {
            v8f acc = {};
#pragma unroll
            for (int s = 0; s < 2; ++s) {
                const int kb = 4 * s;
                v2f a, b;
                a[0] = arow[kb + kk];
                a[1] = arow[kb + kk + 1];
                const int col = t * 16 + mrow;
                b[0] = W1[(kb + kk)     * 64 + col] * s1;
                b[1] = W1[(kb + kk + 1) * 64 + col] * s1;
                acc = __builtin_amdgcn_wmma_f32_16x16x4_f32(
                          false, a, false, b, (short)0, acc, false, false);
            }
#pragma unroll
            for (int r = 0; r < 8; ++r)
                h_lds[wave][r + 8 * half][t * 16 + mrow] = fast_tanh(acc[r]);
        }
    }
    __syncthreads();

    // ---------------- Layer 2: (16x64) @ (64x64), tanh -> h_lds -----------
    {
        float h2[4][8];
#pragma unroll
        for (int t = 0; t < 4; ++t) {
            v8f acc = {};
#pragma unroll
            for (int ks = 0; ks < 16; ++ks) {
                const int kb = ks * 4;
                v2f a, b;
                a[0] = h_lds[wave][mrow][kb + kk];
                a[1] = h_lds[wave][mrow][kb + kk + 1];
                const int col = t * 16 + mrow;
                b[0] = W2[(kb + kk)     * 64 + col] * s23;
                b[1] = W2[(kb + kk + 1) * 64 + col] * s23;
                acc = __builtin_amdgcn_wmma_f32_16x16x4_f32(
                          false, a, false, b, (short)0, acc, false, false);
            }
#pragma unroll
            for (int r = 0; r < 8; ++r) h2[t][r] = fast_tanh(acc[r]);
        }
        __syncthreads();   // everyone done reading h1 before overwrite
#pragma unroll
        for (int t = 0; t < 4; ++t)
#pragma unroll
            for (int r = 0; r < 8; ++r)
                h_lds[wave][r + 8 * half][t * 16 + mrow] = h2[t][r];
    }
    __syncthreads();

    // ---------------- Layer 3: (16x64) @ (64x160) -> w_lds ----------------
    {
#pragma unroll 2
        for (int t = 0; t < 10; ++t) {
            v8f acc = {};
#pragma unroll
            for (int ks = 0; ks < 16; ++ks) {
                const int kb = ks * 4;
                v2f a, b;
                a[0] = h_lds[wave][mrow][kb + kk];
                a[1] = h_lds[wave][mrow][kb + kk + 1];
                const int col = t * 16 + mrow;
                b[0] = W3[(kb + kk)     * W_NUMEL + col] * s23;
                b[1] = W3[(kb + kk + 1) * W_NUMEL + col] * s23;
                acc = __builtin_amdgcn_wmma_f32_16x16x4_f32(
                          false, a, false, b, (short)0, acc, false, false);
            }
#pragma unroll
            for (int r = 0; r < 8; ++r)
                w_lds[wave][r + 8 * half][t * 16 + mrow] = acc[r];
        }
    }
    __syncthreads();

    // ------------- Phase 2: tensor-product message + scatter --------------
    const float inv_den = 1.0f / denom[0];
    const int c = lane;                       // channel, MUL == wave32
    for (int em = 0; em < EDGES_PER_WAVE; ++em) {
        const int e   = e0 + em;
        const int src = eidx[2 * e + 1];
        const int dst = eidx[2 * e + 0];
        const float y0  = esph[4 * e + 0];
        const float y1x = esph[4 * e + 1];
        const float y1y = esph[4 * e + 2];
        const float y1z = esph[4 * e + 3];

        const float* xp = node + (size_t)src * NODE_DIM;
        const float x0  = xp[c];
        const float x1x = xp[MULCH + 3 * c + 0];
        const float x1y = xp[MULCH + 3 * c + 1];
        const float x1z = xp[MULCH + 3 * c + 2];

        const float w0 = w_lds[wave][em][0 * MULCH + c];
        const float w1 = w_lds[wave][em][1 * MULCH + c];
        const float w2 = w_lds[wave][em][2 * MULCH + c];
        const float w3 = w_lds[wave][em][3 * MULCH + c];
        const float w4 = w_lds[wave][em][4 * MULCH + c];

        const float ms0  = x0 * y0 * w0;
        const float dot  = x1x * y1x + x1y * y1y + x1z * y1z;
        const float ms1  = dot * INV_SQRT3F * w1;
        const float mv0x = x0 * y1x * w2;
        const float mv0y = x0 * y1y * w2;
        const float mv0z = x0 * y1z * w2;
        const float mv1x = x1x * y0 * w3;
        const float mv1y = x1y * y0 * w3;
        const float mv1z = x1z * y0 * w3;
        const float crx  = x1y * y1z - x1z * y1y;
        const float cry  = x1z * y1x - x1x * y1z;
        const float crz  = x1x * y1y - x1y * y1x;
        const float mv2x = crx * INV_SQRT2F * w4;
        const float mv2y = cry * INV_SQRT2F * w4;
        const float mv2z = crz * INV_SQRT2F * w4;

        float* mp = msg + (size_t)e * MID_DIM;
        mp[c]                 = ms0;
        mp[32 + c]            = ms1;
        mp[64  + 3 * c + 0]   = mv0x;
        mp[64  + 3 * c + 1]   = mv0y;
        mp[64  + 3 * c + 2]   = mv0z;
        mp[160 + 3 * c + 0]   = mv1x;
        mp[160 + 3 * c + 1]   = mv1y;
        mp[160 + 3 * c + 2]   = mv1z;
        mp[256 + 3 * c + 0]   = mv2x;
        mp[256 + 3 * c + 1]   = mv2y;
        mp[256 + 3 * c + 2]   = mv2z;

        float* op = out + (size_t)dst * MID_DIM;
        atomicAdd(&op[c],               ms0  * inv_den);
        atomicAdd(&op[32 + c],          ms1  * inv_den);
        atomicAdd(&op[64  + 3 * c + 0], mv0x * inv_den);
        atomicAdd(&op[64  + 3 * c + 1], mv0y * inv_den);
        atomicAdd(&op[64  + 3 * c + 2], mv0z * inv_den);
        atomicAdd(&op[160 + 3 * c + 0], mv1x * inv_den);
        atomicAdd(&op[160 + 3 * c + 1], mv1y * inv_den);
        atomicAdd(&op[160 + 3 * c + 2], mv1z * inv_den);
        atomicAdd(&op[256 + 3 * c + 0], mv2x * inv_den);
        atomicAdd(&op[256 + 3 * c + 1], mv2y * inv_den);
        atomicAdd(&op[256 + 3 * c + 2], mv2z * inv_den);
    }
}

extern "C" void kernel_launch(void* const* d_in, const int* in_sizes, int n_in,
                              void* d_out, int out_size, void* d_ws, size_t ws_size,
                              hipStream_t stream) {
    const float* node  = (const float*)d_in[0];
    const float* esph  = (const float*)d_in[1];
    const float* erbf  = (const float*)d_in[2];
    const int*   eidx  = (const int*)  d_in[3];
    const float* W1    = (const float*)d_in[4];
    const float* W2    = (const float*)d_in[5];
    const float* W3    = (const float*)d_in[6];
    const float* denom = (const float*)d_in[7];

    float* out = (float*)d_out;                       // (NNODES, 352)
    float* msg = out + (size_t)NNODES * MID_DIM;      // (NEDGES, 352)

    // zero the segment-sum accumulator region (d_out is poisoned)
    const int n4 = (NNODES * MID_DIM) / 4;
    zero_out_region<<<(n4 + 255) / 256, 256, 0, stream>>>((float4*)out, n4);

    const int blocks = NEDGES / (WAVES_PER_BLOCK * EDGES_PER_WAVE); // 8192
    irreps_conv_fused<<<blocks, WAVES_PER_BLOCK * 32, 0, stream>>>(
        node, esph, erbf, eidx, W1, W2, W3, denom, out, msg);
}